// SSIMLoss_8650064134151
// MI455X (gfx1250) — compile-verified
//
#include <hip/hip_runtime.h>

typedef __attribute__((ext_vector_type(8))) float v8f;
typedef __attribute__((ext_vector_type(2))) float v2f;

#define IMG_H 512
#define IMG_W 512
#define NIMG 32
#define SLAB_COLS 32
#define SLAB_ROWS 64
#define CSLABS (IMG_W / SLAB_COLS)           // 16
#define RSLABS (IMG_H / SLAB_ROWS)           // 8
#define WG_COUNT (NIMG * CSLABS * RSLABS)    // 4096
#define TILE_ROWS 72                          // padded slab rows (need 70, +2 slack)
#define TILE_COLS 40                          // padded slab cols (32 + 2*3 halo + 2 slack)
#define TILE_ELEMS (TILE_ROWS * TILE_COLS)   // 2880
#define LSTRIDE 41                            // odd stride -> conflict-free column walks
#define N_PIX ((float)(NIMG * IMG_H * IMG_W))

#if defined(__HIP_DEVICE_COMPILE__) && __has_builtin(__builtin_amdgcn_global_load_async_to_lds_b32)
#define HAVE_ASYNC_LDS 1
#else
#define HAVE_ASYNC_LDS 0
#endif

typedef __attribute__((address_space(1))) const float gf32;
typedef __attribute__((address_space(1))) int gi32;
typedef __attribute__((address_space(3))) float lf32;
typedef __attribute__((address_space(3))) int li32;

__device__ __forceinline__ float bandv(int k, int n) {
    // ones-band: Bband[k][n] = 1 iff n <= k <= n+6
    return ((unsigned)(k - n) <= 6u) ? 1.0f : 0.0f;
}

__device__ __forceinline__ void wait_async_lds() {
#if __has_builtin(__builtin_amdgcn_s_wait_asynccnt)
    __builtin_amdgcn_s_wait_asynccnt(0);
#else
    asm volatile("s_wait_asynccnt 0x0" ::: "memory");
#endif
}

__global__ __launch_bounds__(256) void ssim_tile_kernel(
    const float* __restrict__ img1, const float* __restrict__ img2,
    float* __restrict__ partial)
{
    // WG-shared padded input slab + per-wave horizontal-sum scratch
    __shared__ float lX[TILE_ROWS][LSTRIDE];
    __shared__ float lY[TILE_ROWS][LSTRIDE];
    __shared__ float lH[8][32][17];
    __shared__ float wsum[8];

    const int tid  = threadIdx.x;
    const int wave = tid >> 5;
    const int lane = tid & 31;
    const int half = lane >> 4;
    const int l16  = lane & 15;

    // block -> (image, row-slab, col-slab)
    const int bid   = blockIdx.x;
    const int img   = bid / (RSLABS * CSLABS);
    const int rem   = bid % (RSLABS * CSLABS);
    const int r0    = (rem / CSLABS) * SLAB_ROWS;
    const int c0    = (rem % CSLABS) * SLAB_COLS;

    // wave -> 16x16 tile inside the slab
    const int ctile = wave & 1;           // 0..1
    const int rtile = wave >> 1;          // 0..3

    const float* p1 = img1 + (size_t)img * (IMG_H * IMG_W);
    const float* p2 = img2 + (size_t)img * (IMG_H * IMG_W);

    // ---- stage padded slab (rows r0-3..r0+68, cols c0-3..c0+36) into LDS ----
    const bool interior = (r0 >= 3) && (r0 + 68 < IMG_H) && (c0 >= 3) && (c0 + 36 < IMG_W);
#if HAVE_ASYNC_LDS
    if (interior) {
        gf32* g1 = (gf32*)(p1 + (size_t)(r0 - 3) * IMG_W + (c0 - 3));
        gf32* g2 = (gf32*)(p2 + (size_t)(r0 - 3) * IMG_W + (c0 - 3));
        #pragma unroll
        for (int i = 0; i < 12; ++i) {
            int idx = tid + 256 * i;
            if (idx < TILE_ELEMS) {
                int rr = idx / TILE_COLS;
                int cc = idx % TILE_COLS;
                size_t go = (size_t)rr * IMG_W + cc;
                __builtin_amdgcn_global_load_async_to_lds_b32(
                    (gi32*)(g1 + go), (li32*)(lf32*)&lX[rr][cc], 0, 0);
                __builtin_amdgcn_global_load_async_to_lds_b32(
                    (gi32*)(g2 + go), (li32*)(lf32*)&lY[rr][cc], 0, 0);
            }
        }
        wait_async_lds();
    } else
#endif
    {
        #pragma unroll
        for (int i = 0; i < 12; ++i) {
            int idx = tid + 256 * i;
            if (idx < TILE_ELEMS) {
                int rr = idx / TILE_COLS;
                int cc = idx % TILE_COLS;
                int gr = r0 - 3 + rr;
                int gc = c0 - 3 + cc;
                bool ok = (gr >= 0) & (gr < IMG_H) & (gc >= 0) & (gc < IMG_W);
                size_t off = (size_t)gr * IMG_W + gc;
                lX[rr][cc] = ok ? p1[off] : 0.0f;
                lY[rr][cc] = ok ? p2[off] : 0.0f;
            }
        }
    }
    __syncthreads();

    float filt[5][8];   // 5 box-filtered channels; lane -> col=l16, rows=8*half+0..7

    #pragma unroll
    for (int ch = 0; ch < 5; ++ch) {
        // ---- horizontal 7-tap as banded GEMM, V_WMMA_F32_16X16X4_F32 chains ----
        #pragma unroll
        for (int s = 0; s < 2; ++s) {
            int row = 16 * rtile + 16 * s + l16;   // padded slab row (A-matrix row)
            row = row > (TILE_ROWS - 1) ? (TILE_ROWS - 1) : row;  // clamp (excess rows unused)
            v8f acc = {};
            #pragma unroll
            for (int kc = 0; kc < 6; ++kc) {
                const int ka = kc * 4 + 2 * half;         // lane's K pair
                const int cb = 16 * ctile + ka;           // column in slab
                float x0 = lX[row][cb], x1 = lX[row][cb + 1];
                float y0 = lY[row][cb], y1 = lY[row][cb + 1];
                v2f a = {};
                switch (ch) {
                    case 0: a.x = x0;      a.y = x1;      break; // img1
                    case 1: a.x = y0;      a.y = y1;      break; // img2
                    case 2: a.x = x0 * x0; a.y = x1 * x1; break; // img1^2
                    case 3: a.x = y0 * y0; a.y = y1 * y1; break; // img2^2
                    case 4: a.x = x0 * y0; a.y = x1 * y1; break; // img1*img2
                }
                v2f b;
                b.x = bandv(ka,     l16);
                b.y = bandv(ka + 1, l16);
                acc = __builtin_amdgcn_wmma_f32_16x16x4_f32(
                    false, a, false, b, (short)0, acc, false, false);
            }
            // D layout: VGPR v, lane -> row = v + 8*half, col = l16
            #pragma unroll
            for (int v = 0; v < 8; ++v)
                lH[wave][16 * s + v + 8 * half][l16] = acc[v];
        }

        // ---- vertical 7-tap, sliding window over LDS-staged H ----
        const int rbase = 8 * half;
        float h[14];
        #pragma unroll
        for (int j = 0; j < 14; ++j) h[j] = lH[wave][rbase + j][l16];
        float s7 = ((h[0] + h[1]) + (h[2] + h[3])) + ((h[4] + h[5]) + h[6]);
        filt[ch][0] = s7;
        #pragma unroll
        for (int i = 1; i < 8; ++i) {
            s7 += h[i + 6] - h[i - 1];
            filt[ch][i] = s7;
        }
        // same-wave LDS ops are in-order (DScnt) -> lH reuse across channels is safe
    }

    // ---- SSIM map + deterministic reduction ----
    const float C1 = 1e-4f;
    const float C2 = 9e-4f;
    float acc = 0.0f;
    #pragma unroll
    for (int i = 0; i < 8; ++i) {
        float mu1 = filt[0][i], mu2 = filt[1][i];
        float e11 = filt[2][i], e22 = filt[3][i], e12 = filt[4][i];
        float s1  = e11 - mu1 * mu1;
        float s2  = e22 - mu2 * mu2;
        float s12 = e12 - mu1 * mu2;
        float num = (2.0f * mu1 * mu2 + C1) * (2.0f * s12 + C2);
        float den = (mu1 * mu1 + mu2 * mu2 + C1) * (s1 + s2 + C2);
        acc += num / den;
    }
    #pragma unroll
    for (int off = 16; off > 0; off >>= 1)
        acc += __shfl_xor(acc, off, 32);
    if (lane == 0) wsum[wave] = acc;
    __syncthreads();
    if (tid == 0) {
        float t = ((wsum[0] + wsum[1]) + (wsum[2] + wsum[3])) +
                  ((wsum[4] + wsum[5]) + (wsum[6] + wsum[7]));
        partial[blockIdx.x] = t;
    }
}

__global__ __launch_bounds__(256) void ssim_reduce_kernel(
    const float* __restrict__ partial, float* __restrict__ out)
{
    __shared__ float sh[256];
    float s = 0.0f;
    for (int i = threadIdx.x; i < WG_COUNT; i += 256) s += partial[i];
    sh[threadIdx.x] = s;
    __syncthreads();
    for (int step = 128; step > 0; step >>= 1) {
        if (threadIdx.x < step) sh[threadIdx.x] += sh[threadIdx.x + step];
        __syncthreads();
    }
    if (threadIdx.x == 0) out[0] = 1.0f - sh[0] / N_PIX;
}

extern "C" void kernel_launch(void* const* d_in, const int* in_sizes, int n_in,
                              void* d_out, int out_size, void* d_ws, size_t ws_size,
                              hipStream_t stream)
{
    const float* img1 = (const float*)d_in[0];
    const float* img2 = (const float*)d_in[1];
    float* out = (float*)d_out;
    float* partial = (float*)d_ws;   // WG_COUNT floats = 16 KB scratch

    ssim_tile_kernel<<<WG_COUNT, 256, 0, stream>>>(img1, img2, partial);
    ssim_reduce_kernel<<<1, 256, 0, stream>>>(partial, out);
}